// GNNmodel_28329604285048
// MI455X (gfx1250) — compile-verified
//
#include <hip/hip_runtime.h>
#include <math.h>

#define N_NODES  100000
#define N_EDGES  1600000
#define N_GRAPHS 64

typedef __attribute__((ext_vector_type(2))) float v2f;
typedef __attribute__((ext_vector_type(8))) float v8f;

__device__ __forceinline__ float silu_f(float x) {
    return x / (1.0f + __expf(-x));
}

// ---------------------------------------------------------------- fills
__global__ void k_fill(float* __restrict__ p, float v, int n) {
    int i = blockIdx.x * blockDim.x + threadIdx.x;
    if (i < n) p[i] = v;
}

// ------------------------------------------------- degree / normalization
__global__ void k_deg(const int* __restrict__ ei, const float* __restrict__ w,
                      float* __restrict__ deg) {
    int e = blockIdx.x * blockDim.x + threadIdx.x;
    if (e < N_EDGES) atomicAdd(deg + ei[N_EDGES + e], w[e]);  // targets
}

__global__ void k_dinv(const float* __restrict__ deg, float* __restrict__ dinv) {
    int i = blockIdx.x * blockDim.x + threadIdx.x;
    if (i < N_NODES) {
        float d = deg[i];
        dinv[i] = d > 0.0f ? rsqrtf(d) : 0.0f;
    }
}

// ---------------------------------------------- layer 1: ts1 = dinv*(x@W1)
__global__ void k_gemm1(const float* __restrict__ x, const float* __restrict__ W1,
                        const float* __restrict__ dinv, float* __restrict__ ts1) {
    int t = blockIdx.x * blockDim.x + threadIdx.x;
    if (t >= N_NODES * 100) return;
    int i = t / 100, f = t % 100;
    float v = x[i * 3 + 0] * W1[f] + x[i * 3 + 1] * W1[100 + f] + x[i * 3 + 2] * W1[200 + f];
    ts1[t] = dinv[i] * v;
}

// ------------------------------------- edge scatter: acc[col] += w * ts[row]
template <int F>
__global__ void k_scatter(const int* __restrict__ ei, const float* __restrict__ w,
                          const float* __restrict__ ts, float* __restrict__ acc) {
    const int NF4 = F / 4;
    long long t = (long long)blockIdx.x * blockDim.x + threadIdx.x;
    if (t >= (long long)N_EDGES * NF4) return;
    int e = (int)(t / NF4);
    int q = (int)(t % NF4);
    int row = ei[e];
    int col = ei[N_EDGES + e];
    float we = w[e];
    float4 v = *(const float4*)(ts + (size_t)row * F + 4 * q);
    float* a = acc + (size_t)col * F + 4 * q;
    atomicAdd(a + 0, we * v.x);
    atomicAdd(a + 1, we * v.y);
    atomicAdd(a + 2, we * v.z);
    atomicAdd(a + 3, we * v.w);
}

// ------------------- h1 = silu(dinv*(acc1 + ts1) + b1), written in place
__global__ void k_finalize1(const float* __restrict__ ts1, const float* __restrict__ dinv,
                            const float* __restrict__ b1, float* __restrict__ acc1) {
    int t = blockIdx.x * blockDim.x + threadIdx.x;
    if (t >= N_NODES * 100) return;
    int i = t / 100, f = t % 100;
    float v = dinv[i] * (acc1[t] + ts1[t]) + b1[f];
    acc1[t] = silu_f(v);
}

// ---------------- layer 2 GEMM via V_WMMA_F32_16X16X4_F32 (fp32 WMMA)
// ts2[i][c] = dinv[i] * sum_k h1[i][k] * W2[k][c];  100000x100 @ 100x200
// one wave = one 16x16 tile; 6250 row tiles x 13 col tiles (200 -> pad 208)
__global__ __launch_bounds__(32) void k_gemm2_wmma(const float* __restrict__ h1,
                                                   const float* __restrict__ W2,
                                                   const float* __restrict__ dinv,
                                                   float* __restrict__ ts2) {
    int tile = blockIdx.x;
    int ct = tile % 13;
    int rt = tile / 13;
    int lane = threadIdx.x;
    int half = lane >> 4;    // 0: K=k,k+1  1: K=k+2,k+3
    int l15  = lane & 15;

    int row  = rt * 16 + l15;                 // A row held by this lane
    int col  = ct * 16 + l15;                 // B/D column held by this lane
    int colc = col < 200 ? col : 199;         // clamp (garbage cols never stored)

    const float* ap = h1 + (size_t)row * 100 + 2 * half;
    const float* bp = W2 + (size_t)(2 * half) * 200 + colc;

    v8f c = {0.f, 0.f, 0.f, 0.f, 0.f, 0.f, 0.f, 0.f};
    for (int k = 0; k < 100; k += 4) {
        v2f a;
        a.x = ap[k];
        a.y = ap[k + 1];
        v2f b;
        b.x = bp[(size_t)k * 200];
        b.y = bp[(size_t)(k + 1) * 200];
        // (neg_a, A, neg_b, B, c_mod, C, reuse_a, reuse_b)
        c = __builtin_amdgcn_wmma_f32_16x16x4_f32(false, a, false, b, (short)0, c,
                                                  false, false);
    }

    // D element r of this lane is [M = rt*16 + 8*half + r][N = l15]
    int mbase = rt * 16 + 8 * half;
    if (col < 200) {
#pragma unroll
        for (int r = 0; r < 8; ++r) {
            float dv = dinv[mbase + r];
            ts2[(size_t)(mbase + r) * 200 + col] = dv * c[r];
        }
    }
}

// ------------- fused finalize layer-2 + mean-pool numerator (atomic psum)
__global__ void k_finalize2_pool(const float* __restrict__ ts2, const float* __restrict__ acc2,
                                 const float* __restrict__ dinv, const float* __restrict__ b2,
                                 const int* __restrict__ batch, float* __restrict__ psum) {
    int t = blockIdx.x * blockDim.x + threadIdx.x;  // node*50 + quad
    if (t >= N_NODES * 50) return;
    int i = t / 50, q = t % 50;
    float dv = dinv[i];
    int g = batch[i];
    int base = i * 200 + 4 * q;
    float4 a = *(const float4*)(acc2 + base);
    float4 s = *(const float4*)(ts2 + base);
    float h0 = silu_f(dv * (a.x + s.x) + b2[4 * q + 0]);
    float h1 = silu_f(dv * (a.y + s.y) + b2[4 * q + 1]);
    float h2 = silu_f(dv * (a.z + s.z) + b2[4 * q + 2]);
    float h3 = silu_f(dv * (a.w + s.w) + b2[4 * q + 3]);
    float* p = psum + g * 200 + 4 * q;
    atomicAdd(p + 0, h0);
    atomicAdd(p + 1, h1);
    atomicAdd(p + 2, h2);
    atomicAdd(p + 3, h3);
}

__global__ void k_count(const int* __restrict__ batch, float* __restrict__ pcnt) {
    int i = blockIdx.x * blockDim.x + threadIdx.x;
    if (i < N_NODES) atomicAdd(pcnt + batch[i], 1.0f);
}

__global__ void k_pooled(const float* __restrict__ psum, const float* __restrict__ pcnt,
                         float* __restrict__ pooled) {
    int t = blockIdx.x * blockDim.x + threadIdx.x;
    if (t >= N_GRAPHS * 200) return;
    int g = t / 200;
    pooled[t] = psum[t] / fmaxf(pcnt[g], 1.0f);
}

__global__ void k_mlp1(const float* __restrict__ pooled, const float* __restrict__ Wl1,
                       const float* __restrict__ bl1, float* __restrict__ h3) {
    int t = blockIdx.x * blockDim.x + threadIdx.x;
    if (t >= N_GRAPHS * 100) return;
    int g = t / 100, j = t % 100;
    float acc = bl1[j];
    const float* p = pooled + g * 200;
#pragma unroll 4
    for (int f = 0; f < 200; ++f) acc += p[f] * Wl1[f * 100 + j];
    h3[t] = silu_f(acc);
}

__global__ void k_out(const float* __restrict__ h3, const float* __restrict__ Wl2,
                      const float* __restrict__ bl2, float* __restrict__ out) {
    int g = threadIdx.x;
    if (g >= N_GRAPHS) return;
    float acc = bl2[0];
    const float* h = h3 + g * 100;
#pragma unroll 4
    for (int j = 0; j < 100; ++j) acc += h[j] * Wl2[j];
    out[g] = acc;
}

extern "C" void kernel_launch(void* const* d_in, const int* in_sizes, int n_in,
                              void* d_out, int out_size, void* d_ws, size_t ws_size,
                              hipStream_t stream) {
    const float* x   = (const float*)d_in[0];
    const float* ea  = (const float*)d_in[1];
    const float* W1  = (const float*)d_in[2];
    const float* b1  = (const float*)d_in[3];
    const float* W2  = (const float*)d_in[4];
    const float* b2  = (const float*)d_in[5];
    const float* Wl1 = (const float*)d_in[6];
    const float* bl1 = (const float*)d_in[7];
    const float* Wl2 = (const float*)d_in[8];
    const float* bl2 = (const float*)d_in[9];
    const int*   ei  = (const int*)d_in[10];
    const int*   bat = (const int*)d_in[11];
    float* out = (float*)d_out;

    // workspace layout (floats) ~241 MB total
    float* ws     = (float*)d_ws;
    float* deg    = ws;                    // 100000
    float* dinv   = deg + N_NODES;         // 100000
    float* ts1    = dinv + N_NODES;        // 10,000,000
    float* acc1   = ts1 + 10000000;        // 10,000,000 (becomes h1 in place)
    float* ts2    = acc1 + 10000000;       // 20,000,000
    float* acc2   = ts2 + 20000000;        // 20,000,000
    float* psum   = acc2 + 20000000;       // 12800
    float* pcnt   = psum + 12800;          // 64
    float* pooled = pcnt + 64;             // 12800
    float* h3     = pooled + 12800;        // 6400

    const int B = 256;
    auto cdiv = [](long long a, long long b) { return (unsigned)((a + b - 1) / b); };

    // init accumulators
    k_fill<<<cdiv(N_NODES, B), B, 0, stream>>>(deg, 1.0f, N_NODES);        // self-loop weight
    k_fill<<<cdiv(10000000, B), B, 0, stream>>>(acc1, 0.0f, 10000000);
    k_fill<<<cdiv(20000000, B), B, 0, stream>>>(acc2, 0.0f, 20000000);
    k_fill<<<cdiv(12800 + 64, B), B, 0, stream>>>(psum, 0.0f, 12800 + 64); // psum+pcnt

    // degree -> dinv
    k_deg<<<cdiv(N_EDGES, B), B, 0, stream>>>(ei, ea, deg);
    k_dinv<<<cdiv(N_NODES, B), B, 0, stream>>>(deg, dinv);

    // layer 1
    k_gemm1<<<cdiv((long long)N_NODES * 100, B), B, 0, stream>>>(x, W1, dinv, ts1);
    k_scatter<100><<<cdiv((long long)N_EDGES * 25, B), B, 0, stream>>>(ei, ea, ts1, acc1);
    k_finalize1<<<cdiv((long long)N_NODES * 100, B), B, 0, stream>>>(ts1, dinv, b1, acc1);

    // layer 2 (WMMA f32 GEMM), then scatter
    k_gemm2_wmma<<<6250 * 13, 32, 0, stream>>>(acc1, W2, dinv, ts2);
    k_scatter<200><<<cdiv((long long)N_EDGES * 50, B), B, 0, stream>>>(ei, ea, ts2, acc2);

    // finalize layer 2 fused with pooling
    k_count<<<cdiv(N_NODES, B), B, 0, stream>>>(bat, pcnt);
    k_finalize2_pool<<<cdiv((long long)N_NODES * 50, B), B, 0, stream>>>(ts2, acc2, dinv, b2,
                                                                         bat, psum);
    k_pooled<<<cdiv(N_GRAPHS * 200, B), B, 0, stream>>>(psum, pcnt, pooled);

    // head MLP
    k_mlp1<<<cdiv(N_GRAPHS * 100, B), B, 0, stream>>>(pooled, Wl1, bl1, h3);
    k_out<<<1, 64, 0, stream>>>(h3, Wl2, bl2, out);
}